// MGAT_32504312496879
// MI455X (gfx1250) — compile-verified
//
#include <hip/hip_runtime.h>

#define NUM_USER 20000
#define NUM_ITEM 20000
#define NN       40000
#define EDGES    250000
#define DIMX     64
#define NEG_SLOPE 0.01f

typedef __attribute__((ext_vector_type(2))) float v2f;
typedef __attribute__((ext_vector_type(8))) float v8f;

__device__ __forceinline__ float leaky(float v) { return v > 0.f ? v : NEG_SLOPE * v; }

// monotonic float -> uint map for atomicMax-based segment_max
__device__ __forceinline__ unsigned fmap(float f) {
    unsigned u = __float_as_uint(f);
    return (u & 0x80000000u) ? ~u : (u | 0x80000000u);
}
__device__ __forceinline__ float funmap(unsigned u) {
    return (u & 0x80000000u) ? __uint_as_float(u & 0x7FFFFFFFu) : __uint_as_float(~u);
}

// src(e) = ei[e]; dst(e) = ei[e+E] (e<E) or ei[e-E] (e>=E)  [doubled edge list]
__device__ __forceinline__ void edge_sd(const int* __restrict__ ei, int e, int& s, int& d) {
    s = ei[e];
    d = (e < EDGES) ? ei[e + EDGES] : ei[e - EDGES];
}

__global__ void fill_zero(float* __restrict__ p, long long n) {
    long long i = (long long)blockIdx.x * blockDim.x + threadIdx.x;
    if (i < n) p[i] = 0.f;
}

__global__ void copy_flat(const float* __restrict__ s, float* __restrict__ d, long long n) {
    long long i = (long long)blockIdx.x * blockDim.x + threadIdx.x;
    if (i < n) d[i] = s[i];
}

// D[M,Nn] = act(A[M,K] @ B[K,Nn] + bias); act: 0=none, 1=tanh, 2=leaky_relu
// One wave32 computes a 16x32 output slab (two 16x16 C tiles sharing the A
// fragment) via V_WMMA_F32_16X16X4_F32. K and LDB are compile-time so B loads
// become immediate-offset clauses and the trip count / remainder are exact.
// K % 4 == 0 for every shape here. OOB rows/cols handled by index clamping:
// clamped (duplicate) A-rows / B-cols only pollute D rows >= M / cols >= Nn,
// which are never stored -> K-loop is branch-free with unconditional loads.
template<int K, int LDB>
__global__ void gemm_f32_wmma(const float* __restrict__ A, int lda,
                              const float* __restrict__ B,
                              const float* __restrict__ bias,
                              float* __restrict__ D, int ldd,
                              int M, int Nn, int act)
{
    int tilesN2 = (Nn + 31) >> 5;       // 32-wide column slabs
    int tilesM  = (M + 15) >> 4;
    int wave = blockIdx.x * (blockDim.x >> 5) + (threadIdx.x >> 5);
    if (wave >= tilesM * tilesN2) return;   // whole-wave exit: EXEC all-1 for WMMA
    int tn = wave % tilesN2, tm = wave / tilesN2;
    int m0 = tm << 4, n0 = tn << 5;
    int lane = threadIdx.x & 31;
    int g = lane >> 4;                  // lane half: supplies K = k + 2*g (+0/+1)
    int r = lane & 15;
    int arow  = min(m0 + r, M - 1);
    int bcol0 = n0 + r;
    int bcol1 = n0 + 16 + r;
    int bc0 = min(bcol0, Nn - 1);
    int bc1 = min(bcol1, Nn - 1);
    const float* Ap  = A + (size_t)arow * lda + (g << 1);  // even elem offset -> 8B aligned
    const float* Bp0 = B + (size_t)(g << 1) * LDB + bc0;
    const float* Bp1 = B + (size_t)(g << 1) * LDB + bc1;

    v8f acc0 = {}, acc1 = {};
    #pragma unroll 4
    for (int k = 0; k < K; k += 4) {
        v2f a = *(const v2f*)Ap;        // global_load_b64, shared by both C tiles
        v2f b0, b1;
        b0.x = Bp0[0];  b0.y = Bp0[LDB];
        b1.x = Bp1[0];  b1.y = Bp1[LDB];
        Ap  += 4;
        Bp0 += 4 * LDB;
        Bp1 += 4 * LDB;
        acc0 = __builtin_amdgcn_wmma_f32_16x16x4_f32(
                   false, a, false, b0, (short)0, acc0, false, false);
        acc1 = __builtin_amdgcn_wmma_f32_16x16x4_f32(
                   false, a, false, b1, (short)0, acc1, false, false);
    }

    bool bok0 = bcol0 < Nn;
    bool bok1 = bcol1 < Nn;
    #pragma unroll
    for (int i = 0; i < 8; i++) {
        int row = m0 + i + (g << 3);
        if (row < M) {
            if (bok0) {
                float v = acc0[i];
                if (bias) v += bias[bcol0];
                if (act == 1) v = tanhf(v);
                else if (act == 2) v = leaky(v);
                D[(size_t)row * ldd + bcol0] = v;
            }
            if (bok1) {
                float v = acc1[i];
                if (bias) v += bias[bcol1];
                if (act == 1) v = tanhf(v);
                else if (act == 2) v = leaky(v);
                D[(size_t)row * ldd + bcol1] = v;
            }
        }
    }
}

__global__ void l2norm_rows(float* __restrict__ x, int rows, int L) {
    int wave = blockIdx.x * (blockDim.x >> 5) + (threadIdx.x >> 5);
    int lane = threadIdx.x & 31;
    if (wave >= rows) return;
    float* p = x + (size_t)wave * L;
    float s = 0.f;
    for (int c = lane; c < L; c += 32) { float v = p[c]; s += v * v; }
    for (int o = 16; o; o >>= 1) s += __shfl_xor(s, o, 32);
    float inv = 1.f / fmaxf(sqrtf(s), 1e-12f);
    for (int c = lane; c < L; c += 32) p[c] *= inv;
}

__global__ void count_deg(const int* __restrict__ ei, float* __restrict__ deg) {
    int e = blockIdx.x * blockDim.x + threadIdx.x;
    if (e >= 2 * EDGES) return;
    atomicAdd(&deg[ei[e]], 1.0f);
}

// wave per edge: inner = <xw[dst], xw[src]>, gate = sigmoid(inner*deg[src]^-0.5),
// logits = inner*gate; atomicMax segment max over dst.
__global__ void edge_logits_kernel(const float* __restrict__ xw, int L,
                                   const int* __restrict__ ei,
                                   const float* __restrict__ deg,
                                   float* __restrict__ logits,
                                   unsigned* __restrict__ segmax)
{
    int wave = blockIdx.x * (blockDim.x >> 5) + (threadIdx.x >> 5);
    int lane = threadIdx.x & 31;
    if (wave >= 2 * EDGES) return;
    int s, d; edge_sd(ei, wave, s, d);
    const float* ps = xw + (size_t)s * L;
    const float* pd = xw + (size_t)d * L;
    float sum = 0.f;
    for (int c = lane; c < L; c += 32) sum += ps[c] * pd[c];
    for (int o = 16; o; o >>= 1) sum += __shfl_xor(sum, o, 32);
    if (lane == 0) {
        float gate = 1.f / (1.f + expf(-sum * rsqrtf(deg[s])));
        float lg = sum * gate;
        logits[wave] = lg;
        atomicMax(&segmax[d], fmap(lg));
    }
}

__global__ void edge_exp_kernel(const int* __restrict__ ei,
                                const float* __restrict__ logits,
                                const unsigned* __restrict__ segmax,
                                float* __restrict__ eexp,
                                float* __restrict__ denom)
{
    int e = blockIdx.x * blockDim.x + threadIdx.x;
    if (e >= 2 * EDGES) return;
    int s, d; edge_sd(ei, e, s, d); (void)s;
    float v = expf(logits[e] - funmap(segmax[d]));
    eexp[e] = v;
    atomicAdd(&denom[d], v);
}

// wave per edge: out[dst] += xw[src] * alpha
__global__ void scatter_kernel(const int* __restrict__ ei,
                               const float* __restrict__ xw, int L,
                               const float* __restrict__ eexp,
                               const float* __restrict__ denom,
                               float* __restrict__ out)
{
    int wave = blockIdx.x * (blockDim.x >> 5) + (threadIdx.x >> 5);
    int lane = threadIdx.x & 31;
    if (wave >= 2 * EDGES) return;
    int s, d; edge_sd(ei, wave, s, d);
    float alpha = eexp[wave] / fmaxf(denom[d], 1e-16f);
    const float* pj = xw + (size_t)s * L;
    float* po = out + (size_t)d * L;
    for (int c = lane; c < L; c += 32) atomicAdd(&po[c], pj[c] * alpha);
}

// h = leaky(l2norm(out + conv_b)), in place over `out`
__global__ void finish_h(float* __restrict__ out, const float* __restrict__ convb, int L) {
    int wave = blockIdx.x * (blockDim.x >> 5) + (threadIdx.x >> 5);
    int lane = threadIdx.x & 31;
    if (wave >= NN) return;
    float* p = out + (size_t)wave * L;
    float s = 0.f;
    for (int c = lane; c < L; c += 32) { float v = p[c] + convb[c]; s += v * v; }
    for (int o = 16; o; o >>= 1) s += __shfl_xor(s, o, 32);
    float inv = 1.f / fmaxf(sqrtf(s), 1e-12f);
    for (int c = lane; c < L; c += 32) p[c] = leaky((p[c] + convb[c]) * inv);
}

__global__ void add_inplace(float* __restrict__ a, const float* __restrict__ b, long long n) {
    long long i = (long long)blockIdx.x * blockDim.x + threadIdx.x;
    if (i < n) a[i] += b[i];
}

// rep += leaky(gout + g_b + xhat)
__global__ void rep_accum(float* __restrict__ rep, const float* __restrict__ gout,
                          const float* __restrict__ gb, const float* __restrict__ xhat)
{
    long long i = (long long)blockIdx.x * blockDim.x + threadIdx.x;
    if (i >= (long long)NN * DIMX) return;
    int c = (int)(i & (DIMX - 1));
    rep[i] += leaky(gout[i] + gb[c] + xhat[i]);
}

// rep holds sum of 3 modality reps; score = <rep[u], rep[it]> / 9
__global__ void score_kernel(const float* __restrict__ rep,
                             const int* __restrict__ users,
                             const int* __restrict__ pos,
                             const int* __restrict__ neg,
                             float* __restrict__ out)
{
    int wave = blockIdx.x * (blockDim.x >> 5) + (threadIdx.x >> 5);
    int lane = threadIdx.x & 31;
    if (wave >= 2048) return;
    int k = wave & 1023;
    int u = users[k];
    int it = (wave < 1024) ? pos[k] : neg[k];
    const float* pu = rep + (size_t)u * DIMX;
    const float* pi = rep + (size_t)it * DIMX;
    float s = pu[lane] * pi[lane] + pu[lane + 32] * pi[lane + 32];
    for (int o = 16; o; o >>= 1) s += __shfl_xor(s, o, 32);
    if (lane == 0) out[wave] = s * (1.0f / 9.0f);
}

static void launch_gemm_t(const float* A, int lda, const float* B, int ldb,
                          const float* bias, float* D, int ldd,
                          int M, int Nn, int K, int act, hipStream_t stream)
{
    int tiles  = ((M + 15) / 16) * ((Nn + 31) / 32);
    int blocks = (tiles + 7) / 8;
#define GEMM_CASE(KK, LL)                                                      \
    if (K == KK && ldb == LL) {                                                \
        gemm_f32_wmma<KK, LL><<<blocks, 256, 0, stream>>>(                     \
            A, lda, B, bias, D, ldd, M, Nn, act);                              \
        return;                                                                \
    }
    GEMM_CASE(2048, 256)
    GEMM_CASE(256, 256)
    GEMM_CASE(256, 64)
    GEMM_CASE(128, 128)
    GEMM_CASE(128, 64)
    GEMM_CASE(100, 100)
    GEMM_CASE(100, 64)
#undef GEMM_CASE
}

extern "C" void kernel_launch(void* const* d_in, const int* in_sizes, int n_in,
                              void* d_out, int out_size, void* d_ws, size_t ws_size,
                              hipStream_t stream)
{
    const int Fd[3] = {2048, 128, 100};
    const int Ld[3] = {256, 128, 100};
    const int LMAX = 256;

    float* ws   = (float*)d_ws;
    float* x    = ws;                                   // [NN, LMAX]
    float* xw   = x   + (size_t)NN * LMAX;              // [NN, LMAX]
    float* outb = xw  + (size_t)NN * LMAX;              // [NN, LMAX] (aggr, reused as h)
    float* xhat = outb + (size_t)NN * LMAX;             // [NN, 64]
    float* gout = xhat + (size_t)NN * DIMX;             // [NN, 64]
    float* rep  = gout + (size_t)NN * DIMX;             // [NN, 64]
    float* deg  = rep  + (size_t)NN * DIMX;             // [NN]
    float* denom = deg + NN;                            // [NN]
    unsigned* segmax = (unsigned*)(denom + NN);         // [NN]
    float* logits = (float*)(segmax + NN);              // [2E]
    float* eexp   = logits + 2 * EDGES;                 // [2E]

    const float* id_emb = (const float*)d_in[30];
    const int* ei       = (const int*)d_in[31];
    const int* users    = (const int*)d_in[32];
    const int* positems = (const int*)d_in[33];
    const int* negitems = (const int*)d_in[34];

    auto zero = [&](float* p, long long n) {
        fill_zero<<<(unsigned)((n + 255) / 256), 256, 0, stream>>>(p, n);
    };

    zero(rep, (long long)NN * DIMX);
    zero(deg, NN);
    count_deg<<<(2 * EDGES + 255) / 256, 256, 0, stream>>>(ei, deg);

    for (int m = 0; m < 3; m++) {
        const float* feat   = (const float*)d_in[m * 10 + 0];
        const float* pref   = (const float*)d_in[m * 10 + 1];
        const float* mlp_w  = (const float*)d_in[m * 10 + 2];
        const float* mlp_b  = (const float*)d_in[m * 10 + 3];
        const float* conv_w = (const float*)d_in[m * 10 + 4];
        const float* conv_b = (const float*)d_in[m * 10 + 5];
        const float* lin_w  = (const float*)d_in[m * 10 + 6];
        const float* lin_b  = (const float*)d_in[m * 10 + 7];
        const float* g_w    = (const float*)d_in[m * 10 + 8];
        const float* g_b    = (const float*)d_in[m * 10 + 9];
        int F = Fd[m], L = Ld[m];

        // x = l2norm(concat(pref, tanh(feat@mlp_w + mlp_b)))
        copy_flat<<<(unsigned)(((long long)NUM_USER * L + 255) / 256), 256, 0, stream>>>(
            pref, x, (long long)NUM_USER * L);
        launch_gemm_t(feat, F, mlp_w, L, mlp_b, x + (size_t)NUM_USER * L, L,
                      NUM_ITEM, L, F, 1, stream);
        l2norm_rows<<<(NN + 7) / 8, 256, 0, stream>>>(x, NN, L);

        // xw = x @ conv_w
        launch_gemm_t(x, L, conv_w, L, nullptr, xw, L, NN, L, L, 0, stream);

        // GAT edge softmax + scatter-add
        zero(denom, NN);
        zero((float*)segmax, NN);          // bit pattern 0 == mapped -inf
        zero(outb, (long long)NN * L);
        edge_logits_kernel<<<(2 * EDGES + 7) / 8, 256, 0, stream>>>(xw, L, ei, deg, logits, segmax);
        edge_exp_kernel<<<(2 * EDGES + 255) / 256, 256, 0, stream>>>(ei, logits, segmax, eexp, denom);
        scatter_kernel<<<(2 * EDGES + 7) / 8, 256, 0, stream>>>(ei, xw, L, eexp, denom, outb);
        finish_h<<<(NN + 7) / 8, 256, 0, stream>>>(outb, conv_b, L);   // outb now holds h

        // xhat = leaky(x@lin_w + lin_b) + id_emb
        launch_gemm_t(x, L, lin_w, DIMX, lin_b, xhat, DIMX, NN, DIMX, L, 2, stream);
        add_inplace<<<(unsigned)(((long long)NN * DIMX + 255) / 256), 256, 0, stream>>>(
            xhat, id_emb, (long long)NN * DIMX);

        // rep += leaky(h@g_w + g_b + xhat)
        launch_gemm_t(outb, L, g_w, DIMX, nullptr, gout, DIMX, NN, DIMX, L, 0, stream);
        rep_accum<<<(unsigned)(((long long)NN * DIMX + 255) / 256), 256, 0, stream>>>(
            rep, gout, g_b, xhat);
    }

    score_kernel<<<(2048 + 7) / 8, 256, 0, stream>>>(rep, users, positems, negitems, (float*)d_out);
}